// SparsePointFeatures_27874337751469
// MI455X (gfx1250) — compile-verified
//
#include <hip/hip_runtime.h>

typedef __attribute__((ext_vector_type(2))) float v2f;
typedef __attribute__((ext_vector_type(4))) float v4f;
typedef __attribute__((ext_vector_type(8))) float v8f;

#define NPTS 64
#define DSTRIDE 65
#define BLK 128
#define DOUT 128

__global__ __launch_bounds__(BLK) void spf_kernel(
    const float* __restrict__ points,      // [B,64,3]
    const int*   __restrict__ mask,        // [B,64]
    const int*   __restrict__ total_cells, // [B]
    const float* __restrict__ rel_w,       // [3,20]
    const float* __restrict__ rel_b,       // [20]
    const float* __restrict__ cdist_w,     // [1,16]
    const float* __restrict__ cdist_b,     // [16]
    const float* __restrict__ count_emb,   // [64,16]
    const float* __restrict__ total_emb,   // [256,24]
    const float* __restrict__ pw_w,        // [5,52]
    const float* __restrict__ pw_b,        // [52]
    float* __restrict__ out)               // [B,64,128]
{
    __shared__ float s_pts[NPTS][4];       // masked points, padded K=4
    __shared__ float s_raw[NPTS][4];       // raw points
    __shared__ float s_sq[NPTS];           // |masked point|^2
    __shared__ float s_m[NPTS];            // mask as float
    __shared__ float s_dist[NPTS][DSTRIDE];
    __shared__ float s_red[BLK];
    __shared__ float s_key[NPTS];
    __shared__ float s_cd[NPTS];
    __shared__ float s_scal[8];            // 0:n 1:max_dist 2..4:centroid 5:cdist_max
    __shared__ float s_out[NPTS * DOUT];   // staged output block (coalesced flush)

    const int b   = blockIdx.x;
    const int tid = threadIdx.x;
    const float INF = __builtin_inff();

    // ---- phase 0: load points / mask ----
    if (tid < NPTS) {
        const float* p = points + ((size_t)b * NPTS + tid) * 3;
        float mx = (float)mask[(size_t)b * NPTS + tid];
        float x = p[0], y = p[1], z = p[2];
        s_raw[tid][0] = x;  s_raw[tid][1] = y;  s_raw[tid][2] = z;  s_raw[tid][3] = 0.f;
        float px = x * mx, py = y * mx, pz = z * mx;
        s_pts[tid][0] = px; s_pts[tid][1] = py; s_pts[tid][2] = pz; s_pts[tid][3] = 0.f;
        s_sq[tid] = px*px + py*py + pz*pz;
        s_m[tid]  = mx;
    }
    __syncthreads();

    // ---- phase 1: Gram = P.P^T via v_wmma_f32_16x16x4_f32, then distances ----
    // A (16x4 f32) layout: lanes 0-15 hold K=0,1 ; lanes 16-31 hold K=2,3.
    // D (16x16 f32) layout: VGPR r -> row r + 8*(lane>>4), col = lane&15.
    const int lane = tid & 31;
    const int half = lane >> 4;
    const int l16  = lane & 15;
    const int iblk = tid >> 5;             // wave id 0..3 -> row block
    float local_max = 0.f;
    {
        const int arow = iblk * 16 + l16;
        v2f a;
        a.x = s_pts[arow][half * 2 + 0];
        a.y = s_pts[arow][half * 2 + 1];
        for (int jblk = 0; jblk < 4; ++jblk) {
            const int bcol = jblk * 16 + l16;
            v2f bb;
            bb.x = s_pts[bcol][half * 2 + 0];
            bb.y = s_pts[bcol][half * 2 + 1];
            v8f c = {};
            c = __builtin_amdgcn_wmma_f32_16x16x4_f32(
                    false, a, false, bb, (short)0, c, false, false);
            const int ncol = jblk * 16 + l16;
            #pragma unroll
            for (int r = 0; r < 8; ++r) {
                const int mrow = iblk * 16 + half * 8 + r;
                float d2 = s_sq[mrow] + s_sq[ncol] - 2.0f * c[r];
                d2 = fmaxf(d2, 0.f);
                float dist = (d2 > 0.f) ? sqrtf(d2) : 0.f;
                s_dist[mrow][ncol] = dist;
                local_max = fmaxf(local_max, dist * (s_m[mrow] * s_m[ncol]));
            }
        }
    }
    s_red[tid] = local_max;
    __syncthreads();

    // ---- serial reductions (cheap, done once per block) ----
    if (tid == 0) {
        float mx = 0.f;
        for (int i = 0; i < BLK; ++i) mx = fmaxf(mx, s_red[i]);
        s_scal[1] = fmaxf(mx, 1e-6f);
    } else if (tid == 32) {
        float n = 0.f, cx = 0.f, cy = 0.f, cz = 0.f;
        for (int i = 0; i < NPTS; ++i) {
            n  += s_m[i];
            cx += s_pts[i][0]; cy += s_pts[i][1]; cz += s_pts[i][2];
        }
        s_scal[0] = n;
        float nv = fmaxf(n, 1.f);
        s_scal[2] = cx / nv; s_scal[3] = cy / nv; s_scal[4] = cz / nv;
    }
    __syncthreads();

    // ---- phase 2: per-point distance stats + centroid-relative ----
    float local_dens = 0.f, mean_d = 0.f, min_d = 0.f, std_d = 0.f;
    float cd = 0.f, relx = 0.f, rely = 0.f, relz = 0.f;
    if (tid < NPTS) {
        const int   i        = tid;
        const float nf       = s_scal[0];
        const float max_dist = s_scal[1];
        const float mi       = s_m[i];
        float t0 = INF, t1 = INF, t2 = INF;     // 3 smallest
        float sum = 0.f, sumsq = 0.f;
        for (int j = 0; j < NPTS; ++j) {
            bool vo = (mi > 0.f) && (s_m[j] > 0.f) && (j != i);
            if (vo) {
                float d = s_dist[i][j];
                sum += d; sumsq += d * d;
                if (d < t0)      { t2 = t1; t1 = t0; t0 = d; }
                else if (d < t1) { t2 = t1; t1 = d; }
                else if (d < t2) { t2 = d; }
            }
        }
        const float nm1      = fmaxf(nf - 1.f, 1.f);
        const float mean_raw = sum / nm1;
        mean_d = mean_raw / max_dist;
        const float k_eff = fminf(3.f, nf - 1.f);
        const float k_c   = fmaxf(k_eff, 1.f);
        float ks = 0.f;
        if (k_eff > 0.f && t0 < INF) ks += t0;
        if (k_eff > 1.f && t1 < INF) ks += t1;
        if (k_eff > 2.f && t2 < INF) ks += t2;
        local_dens = (ks / k_c) / max_dist;
        min_d = (t0 < INF) ? (t0 / max_dist) : 0.f;
        const float cnt = (mi > 0.f) ? (nf - 1.f) : 0.f;
        float var = (sumsq - 2.f * mean_raw * sum + mean_raw * mean_raw * cnt)
                    / fmaxf(nf - 2.f, 1.f);
        float sd = (var > 0.f) ? sqrtf(var) : 0.f;
        std_d = (nf > 2.f) ? (sd / max_dist) : 0.f;
        s_key[i] = (mi > 0.f) ? mean_d : INF;   // rank key (invalid -> +inf)
        relx = s_raw[i][0] - s_scal[2];
        rely = s_raw[i][1] - s_scal[3];
        relz = s_raw[i][2] - s_scal[4];
        cd = sqrtf(fmaxf(relx*relx + rely*rely + relz*relz, 1e-12f));
        s_cd[i] = cd * mi;
    }
    __syncthreads();
    if (tid == 0) {
        float mx = 0.f;
        for (int i = 0; i < NPTS; ++i) mx = fmaxf(mx, s_cd[i]);
        s_scal[5] = fmaxf(mx, 1e-6f);
    }
    __syncthreads();

    // ---- phase 3: rank (stable argsort-argsort) + projections, staged in LDS ----
    if (tid < NPTS) {
        const int   i   = tid;
        const float nf  = s_scal[0];
        const float nm1 = fmaxf(nf - 1.f, 1.f);
        const float ki  = s_key[i];
        int rcnt = 0;
        for (int j = 0; j < NPTS; ++j) {
            float kj = s_key[j];
            if (kj < ki || (kj == ki && j < i)) ++rcnt;
        }
        float rank = (float)rcnt / nm1;
        float f0 = local_dens, f1 = mean_d, f2 = min_d, f3 = std_d, f4 = rank;
        if (!(s_m[i] > 0.f && nf > 1.f)) { f0 = f1 = f2 = f3 = f4 = 0.f; }

        float* op = &s_out[i * DOUT];
        // f_rel (20): rel @ rel_w[3,20] + rel_b
        #pragma unroll
        for (int k = 0; k < 20; ++k)
            op[k] = rel_b[k] + relx * rel_w[k] + rely * rel_w[20 + k] + relz * rel_w[40 + k];
        // f_cdist (16)
        const float cdn = cd / s_scal[5];
        #pragma unroll
        for (int k = 0; k < 16; ++k)
            op[20 + k] = cdn * cdist_w[k] + cdist_b[k];
        // f_count (16)
        int n_idx = (int)fminf(fmaxf(nf, 1.f), 63.f);
        #pragma unroll
        for (int k = 0; k < 16; ++k)
            op[36 + k] = count_emb[n_idx * 16 + k];
        // f_total (24)
        int t = total_cells[b];
        t = t < 0 ? 0 : (t > 255 ? 255 : t);
        #pragma unroll
        for (int k = 0; k < 24; ++k)
            op[52 + k] = total_emb[t * 24 + k];
        // f_pw (52): feats @ pw_w[5,52] + pw_b
        #pragma unroll
        for (int k = 0; k < 52; ++k)
            op[76 + k] = pw_b[k] + f0 * pw_w[k]       + f1 * pw_w[52 + k]
                                 + f2 * pw_w[104 + k] + f3 * pw_w[156 + k]
                                 + f4 * pw_w[208 + k];
    }
    __syncthreads();

    // ---- phase 4: coalesced non-temporal flush of the whole [64,128] block ----
    {
        const v4f* src = (const v4f*)s_out;
        v4f* dst = (v4f*)(out + (size_t)b * NPTS * DOUT);
        #pragma unroll
        for (int q = 0; q < (NPTS * DOUT / 4) / BLK; ++q) {
            int idx = q * BLK + tid;
            __builtin_nontemporal_store(src[idx], &dst[idx]);
        }
    }
}

extern "C" void kernel_launch(void* const* d_in, const int* in_sizes, int n_in,
                              void* d_out, int out_size, void* d_ws, size_t ws_size,
                              hipStream_t stream) {
    const float* points      = (const float*)d_in[0];
    const int*   mask        = (const int*)  d_in[1];
    const int*   total_cells = (const int*)  d_in[2];
    const float* rel_w       = (const float*)d_in[3];
    const float* rel_b       = (const float*)d_in[4];
    const float* cdist_w     = (const float*)d_in[5];
    const float* cdist_b     = (const float*)d_in[6];
    const float* count_emb   = (const float*)d_in[7];
    const float* total_emb   = (const float*)d_in[8];
    const float* pw_w        = (const float*)d_in[9];
    const float* pw_b        = (const float*)d_in[10];
    float*       out         = (float*)d_out;

    const int B = in_sizes[0] / (NPTS * 3);
    spf_kernel<<<B, BLK, 0, stream>>>(points, mask, total_cells,
                                      rel_w, rel_b, cdist_w, cdist_b,
                                      count_emb, total_emb, pw_w, pw_b, out);
}